// GNN_1271310319672
// MI455X (gfx1250) — compile-verified
//
#include <hip/hip_runtime.h>

#define N_NODES 100000
#define N_EDGES 1280000
#define C 64
#define NTILES (N_NODES / 16) // 6250, exact

typedef __attribute__((ext_vector_type(2))) float v2f;
typedef __attribute__((ext_vector_type(8))) float v8f;

__device__ __forceinline__ v8f wmma4(v2f a, v2f b, v8f c) {
  // V_WMMA_F32_16X16X4_F32 : D = A(16x4 f32) * B(4x16 f32) + C(16x16 f32)
  // 8 args: (neg_a, A, neg_b, B, c_mod, C, reuse_a, reuse_b)
  return __builtin_amdgcn_wmma_f32_16x16x4_f32(
      false, a, false, b, (short)0, c, false, false);
}

__global__ void zero_kernel(float* __restrict__ p, int n) {
  int i = blockIdx.x * blockDim.x + threadIdx.x;
  int stride = gridDim.x * blockDim.x;
  for (; i < n; i += stride) p[i] = 0.0f;
}

__global__ void count_kernel(const int* __restrict__ dst,
                             float* __restrict__ cnt, int nE) {
  int e = blockIdx.x * blockDim.x + threadIdx.x;
  if (e < nE) atomicAdd(&cnt[dst[e]], 1.0f);
}

// 16 threads per edge; each thread moves 4 consecutive channels.
__global__ void scatter_kernel(const float* __restrict__ feat,
                               const int* __restrict__ src,
                               const int* __restrict__ dst,
                               float* __restrict__ agg, int nE) {
  int tid = blockIdx.x * blockDim.x + threadIdx.x;
  int e = tid >> 4;
  if (e >= nE) return;
  int c = (tid & 15) << 2;
  int s = src[e];
  int d = dst[e];
  const float4 v = *(const float4*)(feat + (size_t)s * C + c);
  float* p = agg + (size_t)d * C + c;
  atomicAdd(p + 0, v.x);
  atomicAdd(p + 1, v.y);
  atomicAdd(p + 2, v.z);
  atomicAdd(p + 3, v.w);
}

// Fused: out = (agg/max(cnt,1)) @ Wl^T + bias + feat @ Wr^T  (+ optional ReLU)
// One wave handles a 16-node tile; 4 column tiles of 16; K=64 as 16 chunks of 4.
__global__ void sage_kernel(const float* __restrict__ feat,
                            const float* __restrict__ agg,
                            const float* __restrict__ cnt,
                            const float* __restrict__ Wl,
                            const float* __restrict__ bias,
                            const float* __restrict__ Wr,
                            float* __restrict__ out, int relu) {
  const int lane = threadIdx.x & 31;
  const int wave = threadIdx.x >> 5;
  const int tile = blockIdx.x * (blockDim.x >> 5) + wave;
  if (tile >= NTILES) return; // wave-uniform: EXEC stays all-1 for WMMA

  const int l15 = lane & 15;  // row (A) / column (B,C,D) within tile
  const int half = lane >> 4; // K-pair selector per ISA f32 16x4 layout
  const int arow = tile * 16 + l15;

  const float inv = 1.0f / fmaxf(cnt[arow], 1.0f);

  const float* aRow = agg + (size_t)arow * C + half * 2;
  const float* xRow = feat + (size_t)arow * C + half * 2;

  // Preload A operands for all 16 K-chunks (mean-scaled agg + raw feat).
  v2f Aa[16], Ax[16];
#pragma unroll
  for (int k = 0; k < 16; ++k) {
    v2f t = *(const v2f*)(aRow + k * 4);
    Aa[k] = t * inv;
    Ax[k] = *(const v2f*)(xRow + k * 4);
  }

#pragma unroll
  for (int t = 0; t < 4; ++t) {
    // B[k][n] = W[n0+n][k0+k]; lane n(+16*half) loads float2 at
    // W[(n0+n)*64 + k0 + 2*half]  -- same pattern as the A loads.
    const float* wl = Wl + (size_t)(t * 16 + l15) * C + half * 2;
    const float* wr = Wr + (size_t)(t * 16 + l15) * C + half * 2;
    v8f acc0 = {};
    v8f acc1 = {};
#pragma unroll
    for (int k = 0; k < 16; ++k) {
      v2f bl = *(const v2f*)(wl + k * 4);
      v2f br = *(const v2f*)(wr + k * 4);
      acc0 = wmma4(Aa[k], bl, acc0); // independent accumulators ->
      acc1 = wmma4(Ax[k], br, acc1); // back-to-back XDL issue
    }
    const float bv = bias[t * 16 + l15];
    // C/D layout: VGPR v -> row (v + 8*half), col l15
    float* op = out + (size_t)(tile * 16 + half * 8) * C + t * 16 + l15;
#pragma unroll
    for (int v = 0; v < 8; ++v) {
      float val = acc0[v] + acc1[v] + bv;
      if (relu) val = fmaxf(val, 0.0f);
      op[(size_t)v * C] = val;
    }
  }
}

extern "C" void kernel_launch(void* const* d_in, const int* in_sizes, int n_in,
                              void* d_out, int out_size, void* d_ws,
                              size_t ws_size, hipStream_t stream) {
  const float* x   = (const float*)d_in[0];
  const int*   ei  = (const int*)d_in[1]; // int32 (JAX default canonicalization)
  const float* W1l = (const float*)d_in[2];
  const float* b1  = (const float*)d_in[3];
  const float* W1r = (const float*)d_in[4];
  const float* W2l = (const float*)d_in[5];
  const float* b2  = (const float*)d_in[6];
  const float* W2r = (const float*)d_in[7];
  float* out = (float*)d_out;

  // Workspace layout: agg [N*C] | cnt [N] | h [N*C]
  float* agg = (float*)d_ws;
  float* cnt = agg + (size_t)N_NODES * C;
  float* h   = cnt + N_NODES;

  const int* src = ei;
  const int* dst = ei + N_EDGES;

  const int aggN = N_NODES * C;
  const int scatterBlocks = (N_EDGES * 16 + 255) / 256;
  const int sageBlocks = (NTILES + 7) / 8; // 8 waves per 256-thread block

  // ---- Layer 1 ----
  zero_kernel<<<2048, 256, 0, stream>>>(agg, aggN + N_NODES); // agg + cnt
  count_kernel<<<(N_EDGES + 255) / 256, 256, 0, stream>>>(dst, cnt, N_EDGES);
  scatter_kernel<<<scatterBlocks, 256, 0, stream>>>(x, src, dst, agg, N_EDGES);
  sage_kernel<<<sageBlocks, 256, 0, stream>>>(x, agg, cnt, W1l, b1, W1r, h, 1);

  // ---- Layer 2 ----
  zero_kernel<<<2048, 256, 0, stream>>>(agg, aggN);
  scatter_kernel<<<scatterBlocks, 256, 0, stream>>>(h, src, dst, agg, N_EDGES);
  sage_kernel<<<sageBlocks, 256, 0, stream>>>(h, agg, cnt, W2l, b2, W2r, out, 0);
}